// HouseholderReflection_56538949484837
// MI455X (gfx1250) — compile-verified
//
#include <hip/hip_runtime.h>

#define DIM 4096
#define R   8

typedef __attribute__((ext_vector_type(2))) float v2f;
typedef __attribute__((ext_vector_type(8))) float v8f;

// ---------------------------------------------------------------------------
// Kernel 1: modified Gram-Schmidt on hra_u (4096 x 8), writes row-major U.
// One block, 256 threads; each thread owns 16 strided rows.
// ---------------------------------------------------------------------------
__global__ void hh_gram_schmidt(const float* __restrict__ hra_u,
                                float* __restrict__ U) {
    __shared__ float red[256];
    const int t = threadIdx.x;

    float u[16];
    for (int col = 0; col < R; ++col) {
        // load column `col`
        #pragma unroll
        for (int e = 0; e < 16; ++e)
            u[e] = hra_u[(size_t)(t + 256 * e) * R + col];

        // orthogonalize against previously written columns (MGS)
        for (int j = 0; j < col; ++j) {
            float p = 0.0f;
            #pragma unroll
            for (int e = 0; e < 16; ++e)
                p += U[(size_t)(t + 256 * e) * R + j] * u[e];
            red[t] = p;
            __syncthreads();
            for (int s = 128; s > 0; s >>= 1) {
                if (t < s) red[t] += red[t + s];
                __syncthreads();
            }
            const float d = red[0];
            __syncthreads();
            #pragma unroll
            for (int e = 0; e < 16; ++e)
                u[e] -= d * U[(size_t)(t + 256 * e) * R + j];
        }

        // normalize
        float p = 0.0f;
        #pragma unroll
        for (int e = 0; e < 16; ++e) p += u[e] * u[e];
        red[t] = p;
        __syncthreads();
        for (int s = 128; s > 0; s >>= 1) {
            if (t < s) red[t] += red[t + s];
            __syncthreads();
        }
        const float inv = 1.0f / (sqrtf(red[0]) + 1e-6f);
        __syncthreads();
        #pragma unroll
        for (int e = 0; e < 16; ++e)
            U[(size_t)(t + 256 * e) * R + col] = u[e] * inv;
        __syncthreads();   // next column must see the finished U[:, col]
    }
}

// ---------------------------------------------------------------------------
// Kernel 2: WU2[i][k] = -2 * sum_j W[i][j] * U[j][k].
// One block (256 threads) per row; float4 loads of W; LDS reduce 8 partials.
// This pass streams W once, warming L2 (192 MB) for kernel 3's re-read.
// ---------------------------------------------------------------------------
__global__ void hh_compute_wu(const float* __restrict__ W,
                              const float* __restrict__ U,
                              float* __restrict__ WU2) {
    __shared__ float red[R][256];
    const int row = blockIdx.x;
    const int t   = threadIdx.x;

    float acc[R];
    #pragma unroll
    for (int k = 0; k < R; ++k) acc[k] = 0.0f;

    const float* wrow = W + (size_t)row * DIM;
    #pragma unroll
    for (int chunk = 0; chunk < 4; ++chunk) {
        const int j = chunk * 1024 + t * 4;
        const float4 w = *(const float4*)(wrow + j);
        const float* u0 = U + (size_t)j * R;
        #pragma unroll
        for (int k = 0; k < R; ++k)
            acc[k] += w.x * u0[k] + w.y * u0[R + k]
                    + w.z * u0[2 * R + k] + w.w * u0[3 * R + k];
    }

    #pragma unroll
    for (int k = 0; k < R; ++k) red[k][t] = acc[k];
    __syncthreads();
    for (int s = 128; s > 0; s >>= 1) {
        if (t < s) {
            #pragma unroll
            for (int k = 0; k < R; ++k) red[k][t] += red[k][t + s];
        }
        __syncthreads();
    }
    if (t < R) WU2[(size_t)row * R + t] = -2.0f * red[t][0];
}

// ---------------------------------------------------------------------------
// Kernel 3: out = W + WU2 @ U^T via V_WMMA_F32_16X16X4_F32 (K=8 -> 2 WMMAs).
// One wave per 16x16 tile. Block = 256 threads = 8 waves = 8 tile columns.
// A (16x4 fp32): lane m=lane&15, VGPR0/1 hold K = 2*(lane>>4) + {0,1}.
// B (4x16 fp32): lane n=lane&15, VGPR0/1 hold K = 2*(lane>>4) + {0,1}.
// C/D (16x16 fp32): VGPR v holds M = v + 8*(lane>>4), N = lane&15.
// ---------------------------------------------------------------------------
__global__ void hh_apply_rank8(const float* __restrict__ W,
                               const float* __restrict__ U,
                               const float* __restrict__ WU2,
                               float* __restrict__ out) {
    const int lane  = threadIdx.x & 31;
    const int wave  = threadIdx.x >> 5;            // 0..7
    const int tileM = blockIdx.x;                  // 0..255
    const int tileN = blockIdx.y * 8 + wave;       // 0..255
    const int row0  = tileM * 16;
    const int col0  = tileN * 16;
    const int half  = lane >> 4;                   // 0 or 1
    const int l15   = lane & 15;

    // A operands: rows of WU2 (pre-scaled by -2)
    const float* wuRow = WU2 + (size_t)(row0 + l15) * R;
    v2f a0, a1;
    a0.x = wuRow[2 * half + 0];
    a0.y = wuRow[2 * half + 1];
    a1.x = wuRow[4 + 2 * half + 0];
    a1.y = wuRow[4 + 2 * half + 1];

    // B operands: U^T tile -> B[k][n] = U[col0 + n][k]
    const float* uRow = U + (size_t)(col0 + l15) * R;
    v2f b0, b1;
    b0.x = uRow[2 * half + 0];
    b0.y = uRow[2 * half + 1];
    b1.x = uRow[4 + 2 * half + 0];
    b1.y = uRow[4 + 2 * half + 1];

    // C accumulator: the W tile itself
    v8f c;
    #pragma unroll
    for (int v = 0; v < 8; ++v)
        c[v] = W[(size_t)(row0 + v + 8 * half) * DIM + col0 + l15];

    // D = A1*B1 + (A0*B0 + C)  ==  W - 2*(W@U)@U^T  for this tile
    c = __builtin_amdgcn_wmma_f32_16x16x4_f32(false, a0, false, b0,
                                              (short)0, c, false, false);
    c = __builtin_amdgcn_wmma_f32_16x16x4_f32(false, a1, false, b1,
                                              (short)0, c, false, false);

    #pragma unroll
    for (int v = 0; v < 8; ++v)
        out[(size_t)(row0 + v + 8 * half) * DIM + col0 + l15] = c[v];
}

// ---------------------------------------------------------------------------
extern "C" void kernel_launch(void* const* d_in, const int* in_sizes, int n_in,
                              void* d_out, int out_size, void* d_ws, size_t ws_size,
                              hipStream_t stream) {
    const float* W     = (const float*)d_in[0];   // 4096*4096 fp32
    const float* hra_u = (const float*)d_in[1];   // 4096*8   fp32
    float*       out   = (float*)d_out;

    float* U   = (float*)d_ws;                    // 4096*8 fp32 = 128 KB
    float* WU2 = U + (size_t)DIM * R;             // 4096*8 fp32 = 128 KB

    hipLaunchKernelGGL(hh_gram_schmidt, dim3(1), dim3(256), 0, stream,
                       hra_u, U);
    hipLaunchKernelGGL(hh_compute_wu, dim3(DIM), dim3(256), 0, stream,
                       W, U, WU2);
    hipLaunchKernelGGL(hh_apply_rank8, dim3(DIM / 16, 32), dim3(256), 0, stream,
                       W, U, WU2, out);
}